// BatchTopKSAE_2611340116259
// MI455X (gfx1250) — compile-verified
//
#include <hip/hip_runtime.h>

#define DIN   2048
#define DSAE  16384
#define NTOK  4096
#define TOPK  64

typedef __attribute__((ext_vector_type(2))) float v2f;
typedef __attribute__((ext_vector_type(8))) float v8f;

// ---------------------------------------------------------------------------
// Kernel 1: post = relu((x - b_dec) @ W_enc + b_enc), written into the
// feature_acts region of d_out.  WMMA f32 16x16x4, 64x128 block tile,
// k-pair-interleaved B tile (fragment = one ds_load_b64), software-pipelined
// global loads.
// ---------------------------------------------------------------------------
#define TM 64
#define TN 128
#define TK 32
#define LDA 36              // padded A-tile stride (floats)
#define LDBP (2 * TN + 32)  // 288: k-pair-interleaved B-tile stride (floats)

__global__ __launch_bounds__(256) void sae_encode_kernel(
    const float* __restrict__ x, const float* __restrict__ Wenc,
    const float* __restrict__ benc, const float* __restrict__ bdec,
    float* __restrict__ post)
{
    __shared__ float lA[TM * LDA];            //  9.0 KB
    __shared__ float lB[(TK / 2) * LDBP];     // 18.0 KB

    const int tid  = threadIdx.x;
    const int lane = tid & 31;
    const int wave = tid >> 5;
    const int wm   = wave & 3;      // 16-row band within 64
    const int wn   = wave >> 2;     // 64-col band within 128
    const int hh   = lane >> 4;     // lane half
    const int l16  = lane & 15;

    const int tok0 = blockIdx.y * TM;
    const int n0   = blockIdx.x * TN;

    // ---- staging decomposition (per thread) --------------------------------
    // A tile: TM x TK = 512 float4, 2 per thread
    const int a_row0 = (tid + 0)   >> 3, a_c40 = ((tid + 0)   & 7) * 4;
    const int a_row1 = (tid + 256) >> 3, a_c41 = ((tid + 256) & 7) * 4;
    // B tile: 16 k-pairs x 32 col-windows = 512 items, 2 per thread
    const int b_pr0 = (tid + 0)   >> 5, b_c40 = ((tid + 0)   & 31) * 4;
    const int b_pr1 = (tid + 256) >> 5, b_c41 = ((tid + 256) & 31) * 4;

    float4 ra[2], rbE[2], rbO[2];

    auto load_chunk = [&](int kc) {
        float4 xv0 = *(const float4*)(x + (size_t)(tok0 + a_row0) * DIN + kc + a_c40);
        float4 bd0 = *(const float4*)(bdec + kc + a_c40);
        ra[0].x = xv0.x - bd0.x; ra[0].y = xv0.y - bd0.y;
        ra[0].z = xv0.z - bd0.z; ra[0].w = xv0.w - bd0.w;
        float4 xv1 = *(const float4*)(x + (size_t)(tok0 + a_row1) * DIN + kc + a_c41);
        float4 bd1 = *(const float4*)(bdec + kc + a_c41);
        ra[1].x = xv1.x - bd1.x; ra[1].y = xv1.y - bd1.y;
        ra[1].z = xv1.z - bd1.z; ra[1].w = xv1.w - bd1.w;
        rbE[0] = *(const float4*)(Wenc + (size_t)(kc + 2 * b_pr0)     * DSAE + n0 + b_c40);
        rbO[0] = *(const float4*)(Wenc + (size_t)(kc + 2 * b_pr0 + 1) * DSAE + n0 + b_c40);
        rbE[1] = *(const float4*)(Wenc + (size_t)(kc + 2 * b_pr1)     * DSAE + n0 + b_c41);
        rbO[1] = *(const float4*)(Wenc + (size_t)(kc + 2 * b_pr1 + 1) * DSAE + n0 + b_c41);
    };

    v8f c[4] = {};
    load_chunk(0);

    for (int k0 = 0; k0 < DIN; k0 += TK) {
        // ---- commit staged registers to LDS
        {
            float* da0 = lA + a_row0 * LDA + a_c40;
            da0[0] = ra[0].x; da0[1] = ra[0].y; da0[2] = ra[0].z; da0[3] = ra[0].w;
            float* da1 = lA + a_row1 * LDA + a_c41;
            da1[0] = ra[1].x; da1[1] = ra[1].y; da1[2] = ra[1].z; da1[3] = ra[1].w;
            // B: interleave even/odd k rows -> fragment-ready b64 pairs
            float* db0 = lB + b_pr0 * LDBP + b_c40 * 2;
            float4 lo0 = {rbE[0].x, rbO[0].x, rbE[0].y, rbO[0].y};
            float4 hi0 = {rbE[0].z, rbO[0].z, rbE[0].w, rbO[0].w};
            *(float4*)(db0)     = lo0;
            *(float4*)(db0 + 4) = hi0;
            float* db1 = lB + b_pr1 * LDBP + b_c41 * 2;
            float4 lo1 = {rbE[1].x, rbO[1].x, rbE[1].y, rbO[1].y};
            float4 hi1 = {rbE[1].z, rbO[1].z, rbE[1].w, rbO[1].w};
            *(float4*)(db1)     = lo1;
            *(float4*)(db1 + 4) = hi1;
        }
        __syncthreads();

        // ---- prefetch next chunk into registers (overlaps WMMA below)
        if (k0 + TK < DIN) load_chunk(k0 + TK);

        // ---- compute: 8 k-steps x 4 WMMAs
        const float* abase = lA + (wm * 16 + l16) * LDA + 2 * hh;   // A[m][2h + j + kk]
        const float* bbase = lB + hh * LDBP;                        // row kk/2 + hh
        #pragma unroll
        for (int kk = 0; kk < TK; kk += 4) {
            v2f a = *(const v2f*)(abase + kk);
            const float* brow = bbase + (kk >> 1) * LDBP;
            #pragma unroll
            for (int f = 0; f < 4; ++f) {
                int nl = wn * 64 + f * 16 + l16;
                v2f b = *(const v2f*)(brow + nl * 2);
                c[f] = __builtin_amdgcn_wmma_f32_16x16x4_f32(
                    false, a, false, b, (short)0, c[f], false, false);
            }
        }
        __syncthreads();   // guard LDS overwrite by next iteration's stores
    }

    // ---- epilogue: +b_enc, relu, store.  C layout: VGPR r -> M = 8*hh + r
    #pragma unroll
    for (int f = 0; f < 4; ++f) {
        int ng = n0 + wn * 64 + f * 16 + l16;
        float be = benc[ng];
        #pragma unroll
        for (int r = 0; r < 8; ++r) {
            int tok = tok0 + wm * 16 + hh * 8 + r;
            float vv = c[f][r] + be;
            post[(size_t)tok * DSAE + ng] = vv > 0.0f ? vv : 0.0f;
        }
    }
}

// ---------------------------------------------------------------------------
// Kernel 2: exact top-64 per token via 4-pass radix select on float bits
// (all values >= 0 after relu, so uint bit order == float order).
// Zeroes non-kept entries in place; emits deterministic compact (val,idx) list.
// ---------------------------------------------------------------------------
__global__ __launch_bounds__(256) void sae_topk_kernel(
    float* __restrict__ post, float* __restrict__ ovals, int* __restrict__ oidx)
{
    const int tok = blockIdx.x;
    const int tid = threadIdx.x;
    float* row = post + (size_t)tok * DSAE;

    __shared__ unsigned hist[256];
    __shared__ unsigned s_prefix;
    __shared__ int s_remaining;
    __shared__ int sg[256], se[256];

    if (tid == 0) { s_prefix = 0u; s_remaining = TOPK; }
    __syncthreads();

    for (int pass = 0; pass < 4; ++pass) {
        unsigned prefix = s_prefix;
        int remaining   = s_remaining;
        hist[tid] = 0u;
        __syncthreads();
        const int shift = 24 - pass * 8;
        for (int i = 0; i < DSAE / 256; ++i) {
            unsigned u = __float_as_uint(row[i * 256 + tid]);
            bool match = (pass == 0) || ((u >> (shift + 8)) == prefix);
            if (match) atomicAdd(&hist[(u >> shift) & 0xFFu], 1u);
        }
        __syncthreads();
        if (tid == 0) {
            int cum = 0, b = 255;
            for (; b > 0; --b) { cum += (int)hist[b]; if (cum >= remaining) break; }
            if (cum < remaining) { cum += (int)hist[0]; b = 0; }   // falls into bin 0
            s_remaining = remaining - (cum - (int)hist[b]);        // ties still needed
            s_prefix    = (prefix << 8) | (unsigned)b;
        }
        __syncthreads();
    }

    const unsigned T  = s_prefix;        // bits of the k-th largest value
    const int need_eq = s_remaining;     // #elements == T to keep
    const int cnt_gt  = TOPK - need_eq;  // #elements  > T

    // per-thread counts of keepers
    int g = 0, e = 0;
    for (int i = 0; i < DSAE / 256; ++i) {
        unsigned u = __float_as_uint(row[i * 256 + tid]);
        if (u > T) ++g; else if (u == T) ++e;
    }
    sg[tid] = g; se[tid] = e;
    __syncthreads();
    if (tid == 0) {                      // serial exclusive scans (deterministic)
        int ag = 0, ae = 0;
        for (int t = 0; t < 256; ++t) {
            int tg = sg[t]; sg[t] = ag; ag += tg;
            int te = se[t]; se[t] = ae; ae += te;
        }
    }
    __syncthreads();

    int goff = sg[tid], eoff = se[tid];
    float* ov = ovals + (size_t)tok * TOPK;
    int*   oi = oidx  + (size_t)tok * TOPK;
    for (int i = 0; i < DSAE / 256; ++i) {
        int j = i * 256 + tid;
        float f = row[j];
        unsigned u = __float_as_uint(f);
        if (u > T) {
            int slot = goff++;
            ov[slot] = f; oi[slot] = j;
        } else if (u == T) {
            int r = eoff++;
            if (r < need_eq) { int slot = cnt_gt + r; ov[slot] = f; oi[slot] = j; }
            else             { row[j] = 0.0f; }
        } else {
            row[j] = 0.0f;
        }
    }
}

// ---------------------------------------------------------------------------
// Kernel 3: recon[tok] = b_dec + sum_j val_j * W_dec[idx_j, :]
// One WG per token; W_dec (128 MB) is L2-resident on MI455X (192 MB L2).
// ---------------------------------------------------------------------------
__global__ __launch_bounds__(256) void sae_decode_kernel(
    const float* __restrict__ vals, const int* __restrict__ idx,
    const float* __restrict__ Wdec, const float* __restrict__ bdec,
    float* __restrict__ recon)
{
    const int tok = blockIdx.x;
    const int tid = threadIdx.x;
    __shared__ float sv[TOPK];
    __shared__ int   si[TOPK];
    if (tid < TOPK) {
        sv[tid] = vals[(size_t)tok * TOPK + tid];
        si[tid] = idx[(size_t)tok * TOPK + tid];
    }
    __syncthreads();

    float4 acc0 = *(const float4*)(bdec + tid * 4);           // cols [0,1024)
    float4 acc1 = *(const float4*)(bdec + 1024 + tid * 4);    // cols [1024,2048)

    #pragma unroll 4
    for (int j = 0; j < TOPK; ++j) {
        float w = sv[j];
        const float4* wr = (const float4*)(Wdec + (size_t)si[j] * DIN);
        float4 w0 = wr[tid];
        float4 w1 = wr[256 + tid];
        acc0.x = fmaf(w, w0.x, acc0.x); acc0.y = fmaf(w, w0.y, acc0.y);
        acc0.z = fmaf(w, w0.z, acc0.z); acc0.w = fmaf(w, w0.w, acc0.w);
        acc1.x = fmaf(w, w1.x, acc1.x); acc1.y = fmaf(w, w1.y, acc1.y);
        acc1.z = fmaf(w, w1.z, acc1.z); acc1.w = fmaf(w, w1.w, acc1.w);
    }
    float* out = recon + (size_t)tok * DIN;
    *(float4*)(out + tid * 4)        = acc0;
    *(float4*)(out + 1024 + tid * 4) = acc1;
}

// ---------------------------------------------------------------------------
extern "C" void kernel_launch(void* const* d_in, const int* in_sizes, int n_in,
                              void* d_out, int out_size, void* d_ws, size_t ws_size,
                              hipStream_t stream) {
    (void)in_sizes; (void)n_in; (void)out_size; (void)ws_size;
    const float* x    = (const float*)d_in[0];
    const float* Wenc = (const float*)d_in[1];
    const float* benc = (const float*)d_in[2];
    const float* Wdec = (const float*)d_in[3];
    const float* bdec = (const float*)d_in[4];
    // d_in[5] is k (== 64), compiled in as TOPK

    float* out_recon = (float*)d_out;                                // [NTOK, DIN]
    float* out_feat  = out_recon + (size_t)NTOK * DIN;               // [NTOK, DSAE]

    float* ws_vals = (float*)d_ws;                                   // [NTOK, TOPK]
    int*   ws_idx  = (int*)((float*)d_ws + (size_t)NTOK * TOPK);     // [NTOK, TOPK]

    dim3 egrid(DSAE / TN, NTOK / TM);
    sae_encode_kernel<<<egrid, 256, 0, stream>>>(x, Wenc, benc, bdec, out_feat);
    sae_topk_kernel<<<NTOK, 256, 0, stream>>>(out_feat, ws_vals, ws_idx);
    sae_decode_kernel<<<NTOK, 256, 0, stream>>>(ws_vals, ws_idx, Wdec, bdec, out_recon);
}